// LightGCN_71184787964144
// MI455X (gfx1250) — compile-verified
//
#include <hip/hip_runtime.h>
#include <math.h>

#define N_USERS   100000
#define N_ITEMS   50000
#define N_NODES   (N_USERS + N_ITEMS)
#define DIM       64
#define N_LAYERS  3
#define NNZ       4000000
#define BATCH     4096

typedef __attribute__((ext_vector_type(2))) float v2f;
typedef __attribute__((ext_vector_type(8))) float v8f;

// ---------------------------------------------------------------------------
// Kernel 1: build all_emb = concat(user_emb, item_emb) into cur and acc,
// and zero the scatter target nxt. One thread per float element.
// ---------------------------------------------------------------------------
__global__ void lgcn_init(const float* __restrict__ user_emb,
                          const float* __restrict__ item_emb,
                          float* __restrict__ cur,
                          float* __restrict__ acc,
                          float* __restrict__ nxt) {
    size_t i = (size_t)blockIdx.x * blockDim.x + threadIdx.x;
    const size_t total = (size_t)N_NODES * DIM;
    if (i >= total) return;
    const size_t user_elems = (size_t)N_USERS * DIM;
    float v = (i < user_elems) ? user_emb[i] : item_emb[i - user_elems];
    cur[i] = v;
    acc[i] = v;
    nxt[i] = 0.0f;
}

// ---------------------------------------------------------------------------
// Kernel 2: COO SpMM scatter: nxt[row[e]] += vals[e] * cur[col[e]].
// One wave32 per edge; each lane handles 2 consecutive dims (float2), so the
// wave performs one 256B coalesced gather of the column row (L2-resident:
// the whole 38.4MB table fits in the 192MB L2) and 64 f32 atomics that
// resolve in L2. global_prefetch_b8 hides gather latency a few edges ahead.
// ---------------------------------------------------------------------------
#define SPMM_WAVES_PER_BLOCK 8
#define SPMM_LOOKAHEAD 8

__global__ void lgcn_spmm(const int* __restrict__ row,
                          const int* __restrict__ col,
                          const float* __restrict__ vals,
                          const float* __restrict__ cur,
                          float* __restrict__ nxt) {
    int wave = blockIdx.x * SPMM_WAVES_PER_BLOCK + (threadIdx.x >> 5);
    if (wave >= NNZ) return;
    int lane = threadIdx.x & 31;

    int r = row[wave];
    int c = col[wave];
    float v = vals[wave];

    // Prefetch the gather row of an edge a few waves ahead (gfx1250
    // global_prefetch_b8; speculative, no counter impact).
    int pf = wave + SPMM_LOOKAHEAD;
    if (pf < NNZ) {
        const float* pfp = cur + (size_t)col[pf] * DIM + lane * 2;
        __builtin_prefetch(pfp, 0, 0);
    }

    const float2 e = *(const float2*)(cur + (size_t)c * DIM + lane * 2);
    float* dst = nxt + (size_t)r * DIM + lane * 2;
    atomicAdd(dst + 0, v * e.x);
    atomicAdd(dst + 1, v * e.y);
}

// ---------------------------------------------------------------------------
// Kernel 3: layer fold: acc += nxt; cur = nxt; nxt = 0 (ready for next layer).
// ---------------------------------------------------------------------------
__global__ void lgcn_fold(float* __restrict__ cur,
                          float* __restrict__ acc,
                          float* __restrict__ nxt) {
    size_t i = (size_t)blockIdx.x * blockDim.x + threadIdx.x;
    const size_t total = (size_t)N_NODES * DIM;
    if (i >= total) return;
    float t = nxt[i];
    acc[i] += t;
    cur[i] = t;
    nxt[i] = 0.0f;
}

// ---------------------------------------------------------------------------
// Kernel 4: scoring head with V_WMMA_F32_16X16X4_F32.
// One wave (32 threads) handles 16 (user,item) pairs:
//   D = U (16x64) x I^T (64x16) accumulated in f32 over 16 K=4 WMMA steps;
//   scores are the diagonal of D. Diagonal extracted via LDS, then sigmoid.
//
// VGPR layouts per CDNA5 ISA 7.12.2 (32-bit A 16x4): lanes 0-15 carry M=lane
// with {K=k0, K=k0+1} in the two A VGPRs; lanes 16-31 carry {K=k0+2, K=k0+3}.
// B (4x16) mirrors this with N striped across lanes. C/D 16x16 f32: VGPR r,
// lanes 0-15 -> (M=r, N=lane); lanes 16-31 -> (M=r+8, N=lane-16).
// EXEC is all-ones: blockDim.x == 32 and no divergence before the WMMAs.
// ---------------------------------------------------------------------------
__global__ void lgcn_score(const float* __restrict__ acc,
                           const int* __restrict__ users,
                           const int* __restrict__ items,
                           float* __restrict__ out) {
    __shared__ float Ut[16][DIM];   // 16 user rows  (already /4 layer-mean)
    __shared__ float It[16][DIM];   // 16 item rows
    __shared__ float Dt[16][17];    // 16x16 D tile (+1 pad)

    const int lane = threadIdx.x;          // 0..31
    const int tile = blockIdx.x;           // 256 tiles of 16 pairs
    const int rsel = lane & 15;
    const bool isU = lane < 16;
    const int pair = tile * 16 + rsel;

    // Gather one row per lane: lanes 0-15 -> users, lanes 16-31 -> items.
    const int node = isU ? users[pair] : (N_USERS + items[pair]);
    const float4* src = (const float4*)(acc + (size_t)node * DIM);
    float* dstRow = isU ? &Ut[rsel][0] : &It[rsel][0];
    const float scale = 1.0f / (float)(N_LAYERS + 1);   // mean over 4 layers
#pragma unroll
    for (int t = 0; t < DIM / 4; ++t) {
        float4 q = src[t];
        dstRow[4 * t + 0] = q.x * scale;
        dstRow[4 * t + 1] = q.y * scale;
        dstRow[4 * t + 2] = q.z * scale;
        dstRow[4 * t + 3] = q.w * scale;
    }
    __syncthreads();

    // D[m][n] = sum_k U[m][k] * I[n][k], accumulated K=4 at a time.
    const int m  = lane & 15;
    const int kh = (lane >> 4) << 1;       // 0 for lanes 0-15, 2 for 16-31
    v8f cacc = {};
#pragma unroll
    for (int k0 = 0; k0 < DIM; k0 += 4) {
        v2f a, b;
        a.x = Ut[m][k0 + kh + 0];
        a.y = Ut[m][k0 + kh + 1];
        b.x = It[m][k0 + kh + 0];          // B = I^T: B[k][n] = I[n][k]
        b.y = It[m][k0 + kh + 1];
        cacc = __builtin_amdgcn_wmma_f32_16x16x4_f32(
            /*neg_a=*/false, a, /*neg_b=*/false, b,
            /*c_mod=*/(short)0, cacc, /*reuse_a=*/false, /*reuse_b=*/false);
    }

    // Spill the D tile to LDS and pick the diagonal.
    const int rbase = isU ? 0 : 8;
#pragma unroll
    for (int r = 0; r < 8; ++r)
        Dt[rbase + r][m] = cacc[r];
    __syncthreads();

    if (lane < 16) {
        float s = Dt[lane][lane];
        out[tile * 16 + lane] = 1.0f / (1.0f + __expf(-s));
    }
}

// ---------------------------------------------------------------------------
// Launcher. d_in: 0=user_emb 1=item_emb 2=row 3=col 4=vals 5=users 6=items.
// Workspace layout (floats): cur | nxt | acc, each N_NODES*DIM.
// ---------------------------------------------------------------------------
extern "C" void kernel_launch(void* const* d_in, const int* in_sizes, int n_in,
                              void* d_out, int out_size, void* d_ws, size_t ws_size,
                              hipStream_t stream) {
    const float* user_emb = (const float*)d_in[0];
    const float* item_emb = (const float*)d_in[1];
    const int*   row      = (const int*)d_in[2];
    const int*   col      = (const int*)d_in[3];
    const float* vals     = (const float*)d_in[4];
    const int*   users    = (const int*)d_in[5];
    const int*   items    = (const int*)d_in[6];
    float*       out      = (float*)d_out;

    const size_t tableElems = (size_t)N_NODES * DIM;
    float* cur = (float*)d_ws;
    float* nxt = cur + tableElems;
    float* acc = nxt + tableElems;

    const int threads = 256;
    const int gridInit = (int)((tableElems + threads - 1) / threads);
    lgcn_init<<<gridInit, threads, 0, stream>>>(user_emb, item_emb, cur, acc, nxt);

    const int gridSpmm = (NNZ + SPMM_WAVES_PER_BLOCK - 1) / SPMM_WAVES_PER_BLOCK;
    for (int layer = 0; layer < N_LAYERS; ++layer) {
        lgcn_spmm<<<gridSpmm, 32 * SPMM_WAVES_PER_BLOCK, 0, stream>>>(
            row, col, vals, cur, nxt);
        lgcn_fold<<<gridInit, threads, 0, stream>>>(cur, acc, nxt);
    }

    lgcn_score<<<BATCH / 16, 32, 0, stream>>>(acc, users, items, out);
}